// Decoder_18966575579327
// MI455X (gfx1250) — compile-verified
//
#include <hip/hip_runtime.h>

// GRU decoder for MI455X (gfx1250): B=1024, T=2048, V=4, E=16, U=16.
// Strategy: augmented recurrence matmul [h | onehot(tgt)] (16x20) @ Waug via
// V_WMMA_F32_16X16X4_F32 chains; one wave handles a 16-row batch tile over
// the full sequential T loop. Weights live in VGPRs (WMMA A-layout); state
// is kept in WMMA B-layout and rebuilt from the D-layout output each step
// with half-swap shuffles.

typedef float v2f __attribute__((ext_vector_type(2)));
typedef float v8f __attribute__((ext_vector_type(8)));

#define T_LEN 2048

__global__ __launch_bounds__(32) void gru_decoder_wmma(
    const float* __restrict__ h0,      // [B,16] encoder_hidden_state
    const int*   __restrict__ targets, // [B,2048]
    const float* __restrict__ emb,     // [4,16]
    const float* __restrict__ kern,    // [16,48]
    const float* __restrict__ rk,      // [16,48] rec_kernel
    const float* __restrict__ bias,    // [2,48]
    float* __restrict__ out)           // [B,2048,16]
{
    const int lane = threadIdx.x;
    const int li   = lane & 15;   // A: unit index i | B/D: batch row m
    const int hi   = lane >> 4;   // half-wave select
    const int b0   = blockIdx.x * 16;
    const int b    = b0 + li;

    const float* brec = bias + 48;

    // ---- loop-invariant A operands (Waug^T) in WMMA 16x4 A-layout ----
    // A element (M=i, K=j): reg p of chunk c holds j = 4c + p + 2*hi, i = li.
    v2f Az[5], Ar[5], Ah[5];  // gate tiles z, r, hh over K=20 (5 chunks)
    v2f Ax;                   // xh tile: only one-hot chunk (c==4) nonzero
    #pragma unroll
    for (int c = 0; c < 5; ++c) {
        float a[3][2];
        #pragma unroll
        for (int p = 0; p < 2; ++p) {
            if (c < 4) {                      // j = 4c+p+2hi in [0,16): h rows
                int j = 4 * c + p + 2 * hi;
                a[0][p] = rk[j * 48 + li];
                a[1][p] = rk[j * 48 + 16 + li];
                a[2][p] = rk[j * 48 + 32 + li];
            } else {                          // one-hot rows, v = p + 2*hi
                int v = p + 2 * hi;
                float dz = 0.f, dr = 0.f;
                #pragma unroll
                for (int e = 0; e < 16; ++e) {
                    float ev = emb[v * 16 + e];
                    dz += ev * kern[e * 48 + li];
                    dr += ev * kern[e * 48 + 16 + li];
                }
                a[0][p] = dz + bias[li] + brec[li];            // xz + brec_z
                a[1][p] = dr + bias[16 + li] + brec[16 + li];  // xr + brec_r
                a[2][p] = brec[32 + li];                       // b_rec rides one-hot
            }
        }
        Az[c].x = a[0][0]; Az[c].y = a[0][1];
        Ar[c].x = a[1][0]; Ar[c].y = a[1][1];
        Ah[c].x = a[2][0]; Ah[c].y = a[2][1];
    }
    {   // xh table: emb[v]@kern[:,32:48] + bias0
        float d0 = 0.f, d1 = 0.f;
        int v0 = 2 * hi, v1 = 1 + 2 * hi;
        #pragma unroll
        for (int e = 0; e < 16; ++e) {
            d0 += emb[v0 * 16 + e] * kern[e * 48 + 32 + li];
            d1 += emb[v1 * 16 + e] * kern[e * 48 + 32 + li];
        }
        Ax.x = d0 + bias[32 + li];
        Ax.y = d1 + bias[32 + li];
    }

    // ---- state: X[0..3] = h in WMMA B-layout; X[4] = one-hot per step ----
    // B element (K=j, N=m): reg p of chunk c holds j = 4c+p+2hi, m = li.
    v2f X[5];
    float hD[8];  // h in D-layout: reg r holds unit u = r + 8*hi, batch li
    const float* hrow = h0 + (size_t)b * 16;
    #pragma unroll
    for (int c = 0; c < 4; ++c) {
        X[c].x = hrow[4 * c + 0 + 2 * hi];
        X[c].y = hrow[4 * c + 1 + 2 * hi];
    }
    #pragma unroll
    for (int r = 0; r < 8; ++r) hD[r] = hrow[r + 8 * hi];

    const int4* trow = reinterpret_cast<const int4*>(targets + (size_t)b * T_LEN);
    float* orow = out + (size_t)b * T_LEN * 16 + hi * 8;

    const v8f zero = {0.f, 0.f, 0.f, 0.f, 0.f, 0.f, 0.f, 0.f};

    auto sig = [](float x) {
        return __builtin_amdgcn_rcpf(1.0f + __expf(-x));
    };

    auto step = [&](int tgt, int t) {
        // one-hot chunk: element (j=16+p+2hi, m=li) = (tgt[m]==p+2hi)
        X[4].x = (tgt == 2 * hi)     ? 1.0f : 0.0f;
        X[4].y = (tgt == 2 * hi + 1) ? 1.0f : 0.0f;

        v8f Dz = zero, Dr = zero, Dh = zero;
        #pragma unroll
        for (int c = 0; c < 5; ++c) {
            Dz = __builtin_amdgcn_wmma_f32_16x16x4_f32(false, Az[c], false, X[c],
                                                       (short)0, Dz, false, false);
            Dr = __builtin_amdgcn_wmma_f32_16x16x4_f32(false, Ar[c], false, X[c],
                                                       (short)0, Dr, false, false);
            Dh = __builtin_amdgcn_wmma_f32_16x16x4_f32(false, Ah[c], false, X[c],
                                                       (short)0, Dh, false, false);
        }
        v8f Dx = __builtin_amdgcn_wmma_f32_16x16x4_f32(false, Ax, false, X[4],
                                                       (short)0, zero, false, false);

        // gates + state update (D-layout: reg r = unit r+8hi, batch li)
        float hn[8];
        #pragma unroll
        for (int r = 0; r < 8; ++r) {
            float z  = sig(Dz[r]);
            float rr = sig(Dr[r]);
            float cd = sig(Dx[r] + rr * Dh[r]);
            hn[r] = cd + z * (hD[r] - cd);   // z*h + (1-z)*cand
            hD[r] = hn[r];
        }

        // store h_new: units r..r+3 (+8hi) contiguous -> two b128 stores
        float4 lo = {hn[0], hn[1], hn[2], hn[3]};
        float4 hh = {hn[4], hn[5], hn[6], hn[7]};
        *reinterpret_cast<float4*>(orow + (size_t)t * 16)     = lo;
        *reinterpret_cast<float4*>(orow + (size_t)t * 16 + 4) = hh;

        // rebuild B-layout state from D-layout via half-swap shuffles
        float hs[8];
        #pragma unroll
        for (int r = 0; r < 8; ++r) hs[r] = __shfl_xor(hn[r], 16, 32);
        X[0].x = hi ? hs[2] : hn[0];  X[0].y = hi ? hs[3] : hn[1];
        X[1].x = hi ? hs[6] : hn[4];  X[1].y = hi ? hs[7] : hn[5];
        X[2].x = hi ? hn[2] : hs[0];  X[2].y = hi ? hn[3] : hs[1];
        X[3].x = hi ? hn[6] : hs[4];  X[3].y = hi ? hn[7] : hs[5];
    };

    for (int tb = 0; tb < T_LEN; tb += 4) {
        int4 tv = trow[tb >> 2];
        step(tv.x, tb + 0);
        step(tv.y, tb + 1);
        step(tv.z, tb + 2);
        step(tv.w, tb + 3);
    }
}

extern "C" void kernel_launch(void* const* d_in, const int* in_sizes, int n_in,
                              void* d_out, int out_size, void* d_ws, size_t ws_size,
                              hipStream_t stream) {
    const float* h0      = (const float*)d_in[0];  // encoder_hidden_state [1024,16]
    const int*   targets = (const int*)  d_in[1];  // [1024,2048]
    const float* emb     = (const float*)d_in[2];  // [4,16]
    const float* kern    = (const float*)d_in[3];  // [16,48]
    const float* rk      = (const float*)d_in[4];  // [16,48]
    const float* bias    = (const float*)d_in[5];  // [2,48]
    float* out = (float*)d_out;                    // [1024,2048,16]

    // 1024 batch rows / 16 per WMMA tile = 64 waves, one wave per block.
    gru_decoder_wmma<<<64, 32, 0, stream>>>(h0, targets, emb, kern, rk, bias, out);
}